// VQAttention_37821482009266
// MI455X (gfx1250) — compile-verified
//
#include <hip/hip_runtime.h>
#include <hip/hip_bf16.h>

// ---------------------------------------------------------------------------
// VQAttention forward for gfx1250 (MI455X). All GEMMs use V_WMMA_F32_16X16X4_F32
// (exact fp32 tensor op) to match the f32 reference bit-for-bit up to
// accumulation order -- critical because the VQ argmin is discretely sensitive.
// Each wave computes a 32x64 register-blocked output tile (2x4 WMMA tiles).
// ---------------------------------------------------------------------------

typedef float v2f __attribute__((ext_vector_type(2)));
typedef float v8f __attribute__((ext_vector_type(8)));

// Problem dims (fixed by the reference).
constexpr long NT = 16384;  // B*T rows
constexpr long Dd = 1024;
constexpr long Kk = 128;
constexpr long Vv = 1024;
constexpr long Ss = 512;
constexpr long Cc = 512;
constexpr long Rr = 16;
constexpr long Ww = 1024;   // 2*C
constexpr long Ll = 1536;   // 2*C + S
constexpr long Bb = 2;

// ---------------------------------------------------------------------------
// Generic batched f32 WMMA GEMM, 32x64 tile per wave.
//   C[M,N] = A[M,Kd] * op(B) (+ C if beta)       op(B) = B[Kd,N] or B2[N,Kd]^T
// A fragment (16x4 f32): lane l holds (M = l&15, K = k0 + 2*(l>>4) + {0,1}).
// B fragment (4x16 f32): lane l holds (N = l&15, K = k0 + 2*(l>>4) + {0,1}).
// C/D (16x16 f32): lane l holds column N = l&15, VGPR i holds row i + 8*(l>>4).
// skip_mod: if nonzero, batches with (z % skip_mod)==0 are skipped (r==0 cases).
// ---------------------------------------------------------------------------
template <bool TRB>
__global__ __launch_bounds__(32) void gemm_f32_wmma(
    const float* __restrict__ A, long lda, long sA,
    const float* __restrict__ Bm, long ldb, long sB,
    float* __restrict__ Cm, long ldc, long sC,
    int Kd, int beta, int skip_mod)
{
    const int bz = blockIdx.z;
    if (skip_mod != 0 && (bz % skip_mod) == 0) return;
    const float* Ab  = A  + (long)bz * sA;
    const float* Bb2 = Bm + (long)bz * sB;
    float*       Cb  = Cm + (long)bz * sC;

    const int lane = threadIdx.x & 31;
    const int h    = lane >> 4;   // K-half select
    const int lm   = lane & 15;   // M (for A) / N (for B,C,D)
    const long row0 = (long)blockIdx.x * 32;   // 2 M-tiles
    const long col0 = (long)blockIdx.y * 64;   // 4 N-tiles

    v8f acc[2][4];
#pragma unroll
    for (int mt = 0; mt < 2; ++mt)
#pragma unroll
        for (int nt = 0; nt < 4; ++nt)
#pragma unroll
            for (int i = 0; i < 8; ++i)
                acc[mt][nt][i] = beta
                    ? Cb[(row0 + 16 * mt + i + 8 * h) * ldc + col0 + 16 * nt + lm]
                    : 0.0f;

    const float* arow0 = Ab + (row0 + lm) * lda + 2 * h;
    const float* arow1 = arow0 + 16 * lda;
    const float* brow[4];
    if (TRB) {
#pragma unroll
        for (int nt = 0; nt < 4; ++nt)
            brow[nt] = Bb2 + (col0 + 16 * nt + lm) * ldb + 2 * h;
    }

    for (int k0 = 0; k0 < Kd; k0 += 4) {
        v2f a0 = *reinterpret_cast<const v2f*>(arow0 + k0);
        v2f a1 = *reinterpret_cast<const v2f*>(arow1 + k0);
        v2f b[4];
        if (TRB) {
#pragma unroll
            for (int nt = 0; nt < 4; ++nt)
                b[nt] = *reinterpret_cast<const v2f*>(brow[nt] + k0);
        } else {
#pragma unroll
            for (int nt = 0; nt < 4; ++nt) {
                b[nt][0] = Bb2[(long)(k0 + 2 * h)     * ldb + col0 + 16 * nt + lm];
                b[nt][1] = Bb2[(long)(k0 + 2 * h + 1) * ldb + col0 + 16 * nt + lm];
            }
        }
#pragma unroll
        for (int nt = 0; nt < 4; ++nt) {
            acc[0][nt] = __builtin_amdgcn_wmma_f32_16x16x4_f32(
                false, a0, false, b[nt], (short)0, acc[0][nt], false, false);
            acc[1][nt] = __builtin_amdgcn_wmma_f32_16x16x4_f32(
                false, a1, false, b[nt], (short)0, acc[1][nt], false, false);
        }
    }
#pragma unroll
    for (int mt = 0; mt < 2; ++mt)
#pragma unroll
        for (int nt = 0; nt < 4; ++nt)
#pragma unroll
            for (int i = 0; i < 8; ++i)
                Cb[(row0 + 16 * mt + i + 8 * h) * ldc + col0 + 16 * nt + lm] =
                    acc[mt][nt][i];
}

static inline void launch_gemm(hipStream_t st, bool trb,
                               const float* A, long lda, long sA,
                               const float* Bm, long ldb, long sB,
                               float* Cm, long ldc, long sC,
                               long M, long N, int Kd, int beta, int skip_mod,
                               int batch)
{
    dim3 grid((unsigned)(M / 32), (unsigned)(N / 64), (unsigned)batch);
    dim3 blk(32, 1, 1);
    if (trb)
        gemm_f32_wmma<true><<<grid, blk, 0, st>>>(A, lda, sA, Bm, ldb, sB,
                                                  Cm, ldc, sC, Kd, beta, skip_mod);
    else
        gemm_f32_wmma<false><<<grid, blk, 0, st>>>(A, lda, sA, Bm, ldb, sB,
                                                   Cm, ldc, sC, Kd, beta, skip_mod);
}

// --------------------------- elementwise kernels ---------------------------

// RMS-norm: one 256-thread block per token row (D=1024).
__global__ __launch_bounds__(256) void k_rmsnorm(const float* __restrict__ x,
                                                 const float* __restrict__ gsc,
                                                 float* __restrict__ xt)
{
    __shared__ float red[256];
    const long row = blockIdx.x;
    const float* xr = x + row * 1024;
    float s = 0.f;
    for (int j = threadIdx.x; j < 1024; j += 256) { float t = xr[j]; s += t * t; }
    red[threadIdx.x] = s;
    __syncthreads();
    for (int o = 128; o; o >>= 1) {
        if (threadIdx.x < o) red[threadIdx.x] += red[threadIdx.x + o];
        __syncthreads();
    }
    const float rs = rsqrtf(red[0] * (1.0f / 1024.0f) + 1e-6f);
    for (int j = threadIdx.x; j < 1024; j += 256)
        xt[row * 1024 + j] = xr[j] * rs * gsc[j];
}

// q_u = q + xl_u ; q_v = q + xl_v   (NT*K elems, 1/thread)
__global__ __launch_bounds__(256) void k_qbias(const float* __restrict__ q,
                                               const float* __restrict__ xu,
                                               const float* __restrict__ xv,
                                               float* __restrict__ qu,
                                               float* __restrict__ qv)
{
    const long i = (long)blockIdx.x * 256 + threadIdx.x;
    const int j = (int)(i & 127);
    const float qi = q[i];
    qu[i] = qi + xu[j];
    qv[i] = qi + xv[j];
}

// |codebook[s]|^2
__global__ __launch_bounds__(32) void k_cnorm(const float* __restrict__ cbk,
                                              float* __restrict__ cn)
{
    const int s = blockIdx.x, lane = threadIdx.x;
    float acc = 0.f;
    for (int j = lane; j < 128; j += 32) { float c = cbk[s * 128 + j]; acc += c * c; }
    for (int o = 16; o; o >>= 1) acc += __shfl_xor(acc, o, 32);
    if (lane == 0) cn[s] = acc;
}

// VQ nearest-code per token: one wave per token, lane-strided over S=512 codes.
// Ties resolve to lowest index (matches jnp.argmin). Also gathers k_hat.
__global__ __launch_bounds__(256) void k_vq(const float* __restrict__ kk,
                                            const float* __restrict__ cbk,
                                            const float* __restrict__ cn,
                                            int* __restrict__ z,
                                            float* __restrict__ khat)
{
    __shared__ float sk[8][128];
    const int wave = threadIdx.x >> 5, lane = threadIdx.x & 31;
    const long t = (long)blockIdx.x * 8 + wave;
    const float* kr = kk + t * 128;
    for (int j = lane; j < 128; j += 32) sk[wave][j] = kr[j];
    __syncthreads();
    float best = 3.4e38f;
    int bi = 0x7fffffff;
    for (int s = lane; s < 512; s += 32) {
        float d = 0.f;
        const float* cr = cbk + (long)s * 128;
        for (int j = 0; j < 128; ++j) d = fmaf(sk[wave][j], cr[j], d);
        const float val = cn[s] - 2.0f * d;
        if (val < best) { best = val; bi = s; }
    }
    for (int o = 16; o; o >>= 1) {
        float ob = __shfl_xor(best, o, 32);
        int   oi = __shfl_xor(bi, o, 32);
        if (ob < best || (ob == best && oi < bi)) { best = ob; bi = oi; }
    }
    if (lane == 0) z[t] = bi;
    const float* cr = cbk + (long)bi * 128;
    for (int j = lane; j < 128; j += 32) khat[t * 128 + j] = cr[j];
}

// Per (block,code): deterministic ordered accumulation of deltav and delta1.
// blockIdx.x = (b*R + r)*512 + s ; 256 threads each own 4 of V=1024 lanes.
__global__ __launch_bounds__(256) void k_deltav(const int* __restrict__ z,
                                                const float* __restrict__ v,
                                                float* __restrict__ deltav,
                                                float* __restrict__ delta1)
{
    const long idx = blockIdx.x;
    const long blk = idx >> 9;
    const int s = (int)(idx & 511);
    const int* zb = z + blk * 512;
    float acc[4] = {0.f, 0.f, 0.f, 0.f};
    int cnt = 0;
    for (int c = 0; c < 512; ++c) {
        if (zb[c] == s) {
            ++cnt;
            const float* vr = v + (blk * 512 + c) * 1024;
#pragma unroll
            for (int u = 0; u < 4; ++u) acc[u] += vr[threadIdx.x + 256 * u];
        }
    }
    float* dst = deltav + idx * 1024;
#pragma unroll
    for (int u = 0; u < 4; ++u) dst[threadIdx.x + 256 * u] = acc[u];
    if (threadIdx.x == 0) delta1[idx] = (float)cnt;
}

// Serial scan over R=16 blocks producing cache_lower / cache_udl (2-delayed).
// blockIdx.x = b*512 + s ; 256 threads each own 4 of V=1024 lanes.
__global__ __launch_bounds__(256) void k_scan(const float* __restrict__ delta1,
                                              const float* __restrict__ deltav,
                                              float* __restrict__ cache_lower,
                                              float* __restrict__ cache_udl)
{
    const long b = blockIdx.x >> 9;
    const int s = (int)(blockIdx.x & 511);
    float lower = 0.f;
    float udl[4] = {0.f, 0.f, 0.f, 0.f};
    for (int r = 0; r < 2; ++r) {  // cache slots 0,1 are zero
        float* cu = cache_udl + ((b * 16 + r) * 512 + s) * 1024;
#pragma unroll
        for (int u = 0; u < 4; ++u) cu[threadIdx.x + 256 * u] = 0.f;
        if (threadIdx.x == 0) cache_lower[(b * 16 + r) * 512 + s] = 0.f;
    }
    for (int r = 0; r < 16; ++r) {
        const long blk = b * 16 + r;
        const float d1 = delta1[blk * 512 + s];
        const float* dv = deltav + (blk * 512 + s) * 1024;
        const float nl = lower + d1;
        const float inv = 1.f / fmaxf(nl, 1.f);
        const float f1 = lower * inv;
        const float f2 = d1 * inv;
        const float invd = 1.f / fmaxf(d1, 1.f);
#pragma unroll
        for (int u = 0; u < 4; ++u) {
            const float dvn = dv[threadIdx.x + 256 * u] * invd;
            udl[u] = f1 * udl[u] + f2 * dvn;
        }
        lower = nl;
        if (r + 2 < 16) {
            float* cu = cache_udl + ((b * 16 + r + 2) * 512 + s) * 1024;
#pragma unroll
            for (int u = 0; u < 4; ++u) cu[threadIdx.x + 256 * u] = udl[u];
            if (threadIdx.x == 0) cache_lower[(b * 16 + r + 2) * 512 + s] = nl;
        }
    }
}

// Assemble final (pre-softmax) scores in place.
//  cols [0,512)  prev block : (ac + bd[511+j-c])/TAU, -1e30 if r==0
//  cols [512,1024) present  : (ac + bd[1023+jj-c])/TAU, -1e30 if jj>c (causal)
//  cols [1024,1536) cache   : ac/TAU + log(count) (or -1e30 if empty)
__global__ __launch_bounds__(256) void k_assemble(float* __restrict__ scores,
                                                  const float* __restrict__ bd,
                                                  const float* __restrict__ cache_lower)
{
    const long row = blockIdx.x;          // (b*R + r)*C + c
    const int c = (int)(row & 511);
    const long br = row >> 9;
    const int r = (int)(br & 15);
    float* sr = scores + row * 1536;
    const float* bdr = bd + row * 1024;
    const float* clr = cache_lower + br * 512;
    const float TAUi = 0.08838834764831845f;  // 1/sqrt(128)
    for (int j = threadIdx.x; j < 1536; j += 256) {
        float val;
        if (j < 512) {
            val = (r == 0) ? -1e30f : (sr[j] + bdr[511 + j - c]) * TAUi;
        } else if (j < 1024) {
            const int jj = j - 512;
            val = (jj > c) ? -1e30f : (sr[j] + bdr[1023 + jj - c]) * TAUi;
        } else {
            const float cl = clr[j - 1024];
            const float ll = (cl > 0.f) ? logf(fmaxf(cl, 1.f)) : -1e30f;
            val = sr[j] * TAUi + ll;
        }
        sr[j] = val;
    }
}

// Row softmax over 1536 entries, in place.
__global__ __launch_bounds__(256) void k_softmax(float* __restrict__ scores)
{
    __shared__ float red[256];
    const long row = blockIdx.x;
    float* sr = scores + row * 1536;
    float m = -3.4e38f;
    for (int j = threadIdx.x; j < 1536; j += 256) m = fmaxf(m, sr[j]);
    red[threadIdx.x] = m;
    __syncthreads();
    for (int o = 128; o; o >>= 1) {
        if (threadIdx.x < o) red[threadIdx.x] = fmaxf(red[threadIdx.x], red[threadIdx.x + o]);
        __syncthreads();
    }
    m = red[0];
    __syncthreads();
    float s = 0.f;
    for (int j = threadIdx.x; j < 1536; j += 256) {
        const float e = __expf(sr[j] - m);
        sr[j] = e;
        s += e;
    }
    red[threadIdx.x] = s;
    __syncthreads();
    for (int o = 128; o; o >>= 1) {
        if (threadIdx.x < o) red[threadIdx.x] += red[threadIdx.x + o];
        __syncthreads();
    }
    const float inv = 1.f / red[0];
    for (int j = threadIdx.x; j < 1536; j += 256) sr[j] *= inv;
}

// wv *= silu(g)
__global__ __launch_bounds__(256) void k_gate(float* __restrict__ wv,
                                              const float* __restrict__ g)
{
    const long i = (long)blockIdx.x * 256 + threadIdx.x;
    const float gv = g[i];
    wv[i] *= gv / (1.f + __expf(-gv));
}

// ---------------------------------------------------------------------------

extern "C" void kernel_launch(void* const* d_in, const int* in_sizes, int n_in,
                              void* d_out, int out_size, void* d_ws, size_t ws_size,
                              hipStream_t stream)
{
    (void)in_sizes; (void)n_in; (void)out_size; (void)ws_size;
    const float* x    = (const float*)d_in[0];
    const float* ln   = (const float*)d_in[1];
    const float* w_q  = (const float*)d_in[2];
    const float* w_k  = (const float*)d_in[3];
    const float* w_v  = (const float*)d_in[4];
    const float* w_g  = (const float*)d_in[5];
    const float* w_o  = (const float*)d_in[6];
    const float* cbk  = (const float*)d_in[7];
    const float* xl_r = (const float*)d_in[8];
    const float* xl_u = (const float*)d_in[9];
    const float* xl_v = (const float*)d_in[10];

    float* w = (float*)d_ws;
    float* xt        = w; w += NT * Dd;
    float* q         = w; w += NT * Kk;
    float* qu        = w; w += NT * Kk;
    float* qv        = w; w += NT * Kk;
    float* kk        = w; w += NT * Kk;
    float* khat      = w; w += NT * Kk;
    float* v         = w; w += NT * Vv;
    float* g         = w; w += NT * Vv;
    float* deltav    = w; w += Bb * Rr * Ss * Vv;   // later reused as wv
    float* cache_udl = w; w += Bb * Rr * Ss * Vv;
    float* scores    = w; w += NT * Ll;
    float* delta1    = w; w += Bb * Rr * Ss;
    float* cache_lw  = w; w += Bb * Rr * Ss;
    float* cnorm2    = w; w += Ss;
    int*   z         = (int*)w;
    float* bd        = (float*)d_out;   // [NT, W] scratch; dead before final GEMM
    float* wv        = deltav;          // deltav dead after scan

    // 1) RMS-norm + projections
    k_rmsnorm<<<(unsigned)NT, 256, 0, stream>>>(x, ln, xt);
    launch_gemm(stream, false, xt, Dd, 0, w_q, Kk, 0, q,  Kk, 0, NT, Kk, (int)Dd, 0, 0, 1);
    launch_gemm(stream, false, xt, Dd, 0, w_k, Kk, 0, kk, Kk, 0, NT, Kk, (int)Dd, 0, 0, 1);
    launch_gemm(stream, false, xt, Dd, 0, w_v, Vv, 0, v,  Vv, 0, NT, Vv, (int)Dd, 0, 0, 1);
    launch_gemm(stream, false, xt, Dd, 0, w_g, Vv, 0, g,  Vv, 0, NT, Vv, (int)Dd, 0, 0, 1);
    k_qbias<<<(unsigned)(NT * Kk / 256), 256, 0, stream>>>(q, xl_u, xl_v, qu, qv);

    // 2) Vector quantizer
    k_cnorm<<<(unsigned)Ss, 32, 0, stream>>>(cbk, cnorm2);
    k_vq<<<(unsigned)(NT / 8), 256, 0, stream>>>(kk, cbk, cnorm2, z, khat);

    // 3) Cache statistics + serial scan
    k_deltav<<<(unsigned)(Bb * Rr * Ss), 256, 0, stream>>>(z, v, deltav, delta1);
    k_scan<<<(unsigned)(Bb * Ss), 256, 0, stream>>>(delta1, deltav, cache_lw, cache_udl);

    // 4) Scores (raw A*C^T pieces), rel-pos bd, assemble, softmax
    launch_gemm(stream, true, qu, Kk, Cc * Kk, khat,           Kk, Cc * Kk,
                scores + 512,  Ll, Cc * Ll, Cc, Cc, (int)Kk, 0, 0, (int)(Bb * Rr));
    launch_gemm(stream, true, qu, Kk, Cc * Kk, khat - Cc * Kk, Kk, Cc * Kk,
                scores,        Ll, Cc * Ll, Cc, Cc, (int)Kk, 0, (int)Rr, (int)(Bb * Rr));
    launch_gemm(stream, true, qu, Kk, 0,       cbk,            Kk, 0,
                scores + 1024, Ll, 0,       NT, Ss, (int)Kk, 0, 0, 1);
    launch_gemm(stream, true, qv, Kk, 0,       xl_r,           Kk, 0,
                bd,            Ww, 0,       NT, Ww, (int)Kk, 0, 0, 1);
    k_assemble<<<(unsigned)NT, 256, 0, stream>>>(scores, bd, cache_lw);
    k_softmax<<<(unsigned)NT, 256, 0, stream>>>(scores);

    // 5) wv = a_pres@v + a_prev@v_prev + a_cache@cache_udl
    launch_gemm(stream, false, scores + 512,  Ll, Cc * Ll, v,            Vv, Cc * Vv,
                wv, Vv, Cc * Vv, Cc, Vv, (int)Cc, 0, 0,        (int)(Bb * Rr));
    launch_gemm(stream, false, scores,        Ll, Cc * Ll, v - Cc * Vv,  Vv, Cc * Vv,
                wv, Vv, Cc * Vv, Cc, Vv, (int)Cc, 1, (int)Rr,  (int)(Bb * Rr));
    launch_gemm(stream, false, scores + 1024, Ll, Cc * Ll, cache_udl,    Vv, Ss * Vv,
                wv, Vv, Cc * Vv, Cc, Vv, (int)Cc, 1, 0,        (int)(Bb * Rr));

    // 6) Gate + output projection
    k_gate<<<(unsigned)(NT * Vv / 256), 256, 0, stream>>>(wv, g);
    launch_gemm(stream, false, wv, Vv, 0, w_o, Dd, 0, (float*)d_out, Dd, 0,
                NT, Dd, (int)Vv, 0, 0, 1);
}